// GCN_66443144069548
// MI455X (gfx1250) — compile-verified
//
#include <hip/hip_runtime.h>

typedef float v2f __attribute__((ext_vector_type(2)));
typedef float v8f __attribute__((ext_vector_type(8)));

#define CH 128   // IN_CH == HIDDEN == 128 (feature width of every GEMM input)

// ---------------------------------------------------------------------------
// GEMM: C[nrows x NCOL] = A[nrows x CH] @ W[CH x NCOL]  (+ optional bias)
// fp32 WMMA 16x16x4. One wave owns a 16-row strip across all NCOL columns.
// W staged into LDS transposed (K fastest, pitch 130) -> ds_load_b64 per frag.
// ---------------------------------------------------------------------------
template <int NCOL>
__global__ __launch_bounds__(256) void gemm_wmma_f32(
    const float* __restrict__ A, const float* __restrict__ W,
    const float* __restrict__ bias, float* __restrict__ C, int nrows) {
  constexpr int NT = NCOL / 16;
  constexpr int KP = 130;  // K pitch: even (8B align) and 130%64=2 -> no bank conflicts
  __shared__ float lw[NCOL * KP];

  const int tid = threadIdx.x;
  for (int i = tid; i < CH * NCOL; i += 256) {
    int k = i / NCOL, c = i - k * NCOL;
    lw[c * KP + k] = W[i];  // W row-major [K][NCOL] -> lw[col][k]
  }
  __syncthreads();

  const int wave = tid >> 5;
  const int lane = tid & 31;
  const int m0 = (blockIdx.x * 8 + wave) * 16;
  if (m0 >= nrows) return;

  const int lrow = lane & 15;        // A: row-in-tile, B/C: col-in-ntile
  const int kbase = (lane >> 4) * 2; // which K pair this half-wave holds

  v8f acc[NT] = {};

  int arow = m0 + lrow;
  if (arow > nrows - 1) arow = nrows - 1;  // clamp; stores are bounds-checked
  const float* aptr = A + (size_t)arow * CH + kbase;

#pragma unroll 4
  for (int kc = 0; kc < CH / 4; ++kc) {
    float2 av = *(const float2*)(aptr + kc * 4);
    v2f a;
    a[0] = av.x; a[1] = av.y;
#pragma unroll
    for (int n = 0; n < NT; ++n) {
      float2 bv = *(const float2*)&lw[(n * 16 + lrow) * KP + kc * 4 + kbase];
      v2f b;
      b[0] = bv.x; b[1] = bv.y;
      acc[n] = __builtin_amdgcn_wmma_f32_16x16x4_f32(false, a, false, b,
                                                     (short)0, acc[n],
                                                     false, false);
    }
  }

  // C/D layout: lanes 0-15 -> rows m0+r, lanes 16-31 -> rows m0+8+r
  const int rbase = m0 + ((lane >> 4) << 3);
#pragma unroll
  for (int r = 0; r < 8; ++r) {
    int row = rbase + r;
    if (row < nrows) {
#pragma unroll
      for (int n = 0; n < NT; ++n) {
        int col = n * 16 + lrow;
        float v = acc[n][r];
        if (bias) v += bias[col];
        C[(size_t)row * NCOL + col] = v;
      }
    }
  }
}

// ---------------------------------------------------------------------------
// helper kernels
// ---------------------------------------------------------------------------
__global__ void fill_f32(float* __restrict__ p, float v, int n) {
  int i = blockIdx.x * blockDim.x + threadIdx.x;
  if (i < n) p[i] = v;
}

__global__ void zero4(float* __restrict__ p, long n4) {
  long i = (long)blockIdx.x * blockDim.x + threadIdx.x;
  if (i < n4) ((float4*)p)[i] = make_float4(0.f, 0.f, 0.f, 0.f);
}

__global__ void deg_accum(const int* __restrict__ dst, float* __restrict__ deg,
                          int ne) {
  int e = blockIdx.x * blockDim.x + threadIdx.x;
  if (e < ne) atomicAdd(&deg[dst[e]], 1.0f);
}

__global__ void rsqrt_inplace(float* __restrict__ d, int n) {
  int i = blockIdx.x * blockDim.x + threadIdx.x;
  if (i < n) {
    float v = d[i];
    d[i] = v > 0.f ? rsqrtf(v) : 0.f;
  }
}

__global__ void edge_norm(const int* __restrict__ src,
                          const int* __restrict__ dst,
                          const float* __restrict__ invs,
                          float* __restrict__ nrm, int ne) {
  int e = blockIdx.x * blockDim.x + threadIdx.x;
  if (e < ne) nrm[e] = invs[src[e]] * invs[dst[e]];
}

// one wave per edge; lane handles 4 channels (float4) -> 4 f32 atomics
__global__ __launch_bounds__(256) void aggregate(
    const float* __restrict__ t, const int* __restrict__ src,
    const int* __restrict__ dst, const float* __restrict__ nrm,
    float* __restrict__ agg, int ne) {
  const int lane = threadIdx.x & 31;
  int w = (blockIdx.x * blockDim.x + threadIdx.x) >> 5;
  const int nw = (gridDim.x * blockDim.x) >> 5;
  for (int e = w; e < ne; e += nw) {
    const int s = src[e];
    const int d = dst[e];
    const float nv = nrm[e];
    const float4 v = *(const float4*)(t + (size_t)s * CH + lane * 4);
    float* o = agg + (size_t)d * CH + lane * 4;
    atomicAdd(o + 0, v.x * nv);
    atomicAdd(o + 1, v.y * nv);
    atomicAdd(o + 2, v.z * nv);
    atomicAdd(o + 3, v.w * nv);
  }
}

// h = act(agg + t * invs^2 + bias)   (self-loop norm = invs[i]*invs[i])
__global__ void self_bias_act(float* __restrict__ hb,
                              const float* __restrict__ t,
                              const float* __restrict__ invs,
                              const float* __restrict__ bias, int n, int relu) {
  int i = blockIdx.x * blockDim.x + threadIdx.x;  // n*32 threads
  if (i >= n * 32) return;
  int node = i >> 5, c = (i & 31) * 4;
  float s = invs[node];
  s = s * s;
  float4 a = *(float4*)(hb + (size_t)node * CH + c);
  float4 tv = *(const float4*)(t + (size_t)node * CH + c);
  float4 b = *(const float4*)(bias + c);
  float4 r;
  r.x = a.x + tv.x * s + b.x;
  r.y = a.y + tv.y * s + b.y;
  r.z = a.z + tv.z * s + b.z;
  r.w = a.w + tv.w * s + b.w;
  if (relu) {
    r.x = fmaxf(r.x, 0.f);
    r.y = fmaxf(r.y, 0.f);
    r.z = fmaxf(r.z, 0.f);
    r.w = fmaxf(r.w, 0.f);
  }
  *(float4*)(hb + (size_t)node * CH + c) = r;
}

// ---------------------------------------------------------------------------
extern "C" void kernel_launch(void* const* d_in, const int* in_sizes, int n_in,
                              void* d_out, int out_size, void* d_ws,
                              size_t ws_size, hipStream_t stream) {
  const float* x = (const float*)d_in[0];
  const int* ei = (const int*)d_in[1];
  const float* W1 = (const float*)d_in[2];
  const float* b1 = (const float*)d_in[3];
  const float* W2 = (const float*)d_in[4];
  const float* b2 = (const float*)d_in[5];
  const float* W3 = (const float*)d_in[6];
  const float* b3 = (const float*)d_in[7];
  const float* Wo = (const float*)d_in[8];
  const float* bo = (const float*)d_in[9];

  const int N = in_sizes[0] / CH;
  const int E = in_sizes[1] / 2;
  const int* src = ei;
  const int* dstp = ei + E;

  float* bufA = (float*)d_ws;                  // t = h @ W
  float* bufB = bufA + (size_t)N * CH;         // agg / h
  float* invs = bufB + (size_t)N * CH;         // deg -> deg^-1/2
  float* nrm = invs + N;                       // per-edge norm

  const dim3 blk(256);
  const int gN = (N + 255) / 256;
  const int gE = (E + 255) / 256;
  const int gGemm = (N + 127) / 128;           // 8 waves x 16 rows per block
  const long n4 = (long)N * CH / 4;
  const int gZero = (int)((n4 + 255) / 256);
  const int gEpi = (N * 32 + 255) / 256;
  const int gAgg = 8192;                       // grid-stride over edges

  // normalization: deg = 1 (self-loop) + in-degree; invs = deg^-1/2
  fill_f32<<<gN, blk, 0, stream>>>(invs, 1.0f, N);
  deg_accum<<<gE, blk, 0, stream>>>(dstp, invs, E);
  rsqrt_inplace<<<gN, blk, 0, stream>>>(invs, N);
  edge_norm<<<gE, blk, 0, stream>>>(src, dstp, invs, nrm, E);

  // layer 1
  gemm_wmma_f32<128><<<gGemm, blk, 0, stream>>>(x, W1, nullptr, bufA, N);
  zero4<<<gZero, blk, 0, stream>>>(bufB, n4);
  aggregate<<<gAgg, blk, 0, stream>>>(bufA, src, dstp, nrm, bufB, E);
  self_bias_act<<<gEpi, blk, 0, stream>>>(bufB, bufA, invs, b1, N, 1);

  // layer 2
  gemm_wmma_f32<128><<<gGemm, blk, 0, stream>>>(bufB, W2, nullptr, bufA, N);
  zero4<<<gZero, blk, 0, stream>>>(bufB, n4);
  aggregate<<<gAgg, blk, 0, stream>>>(bufA, src, dstp, nrm, bufB, E);
  self_bias_act<<<gEpi, blk, 0, stream>>>(bufB, bufA, invs, b2, N, 1);

  // layer 3 (no relu)
  gemm_wmma_f32<128><<<gGemm, blk, 0, stream>>>(bufB, W3, nullptr, bufA, N);
  zero4<<<gZero, blk, 0, stream>>>(bufB, n4);
  aggregate<<<gAgg, blk, 0, stream>>>(bufA, src, dstp, nrm, bufB, E);
  self_bias_act<<<gEpi, blk, 0, stream>>>(bufB, bufA, invs, b3, N, 0);

  // output projection: [N,128] @ [128,64] + bout -> d_out
  gemm_wmma_f32<64><<<gGemm, blk, 0, stream>>>(bufB, Wo, bo, (float*)d_out, N);
}